// MoleculeAttnBias_90829968376353
// MI455X (gfx1250) — compile-verified
//
#include <hip/hip_runtime.h>
#include <hip/hip_bf16.h>

#define NHEAD 32
#define NVOC  1537
#define NHOP  20
#define NB    16
#define NN    64

typedef __attribute__((ext_vector_type(2))) float v2f;
typedef __attribute__((ext_vector_type(8))) float v8f;

// -----------------------------------------------------------------------------
// Kernel 1: table[d][v][h] = sum_k edge_enc_w[v][k] * Wd[d][k][h]
// 20 small f32 GEMMs (1537x32 @ 32x32) via V_WMMA_F32_16X16X4_F32.
// One 16x16 output tile per wave; 8 k-steps of 4. The ragged last v-tile is
// handled behind wave-uniform branches so full tiles have no EXEC churn.
// -----------------------------------------------------------------------------
__global__ __launch_bounds__(256) void build_table_wmma(
    const float* __restrict__ edge_enc_w,     // [1537][32]
    const float* __restrict__ edge_dis_enc_w, // Wd[d][k][h] = [d*1024 + k*32 + h]
    float* __restrict__ table)                // [20][1537][32]
{
    const int NT_V  = (NVOC + 15) / 16;       // 97 v-tiles (last one has 1 row)
    const int TOTAL = NHOP * NT_V * 2;        // 3880 wave-tiles

    int wave = blockIdx.x * (blockDim.x >> 5) + (threadIdx.x >> 5);
    wave = __builtin_amdgcn_readfirstlane(wave);
    if (wave >= TOTAL) return;                // wave-uniform

    const int ht = wave & 1;                  // h-tile (0..1)
    const int vt = (wave >> 1) % NT_V;        // v-tile (0..96)
    const int d  = (wave >> 1) / NT_V;        // hop    (0..19)
    const bool full_tile = (vt < NT_V - 1);   // wave-uniform

    const int lane = threadIdx.x & 31;
    const int lo   = lane & 15;
    const int kb   = (lane >> 4) << 1;        // 0 or 2 (A/B fragment k-offset)

    const int m  = vt * 16 + lo;              // A row (v index)
    const int mc = m < NVOC ? m : (NVOC - 1);
    const int n  = ht * 16 + lo;              // B/C column (h index)

    const float* A  = edge_enc_w + (size_t)mc * NHEAD;
    const float* Wd = edge_dis_enc_w + (size_t)d * NHEAD * NHEAD;

    v8f c = {0.f, 0.f, 0.f, 0.f, 0.f, 0.f, 0.f, 0.f};

    if (full_tile) {
        // Hot path: no masking at all; pure load -> wmma pipeline.
#pragma unroll
        for (int k0 = 0; k0 < NHEAD; k0 += 4) {
            v2f a, bf;
            a.x  = A[k0 + kb];
            a.y  = A[k0 + kb + 1];
            bf.x = Wd[(k0 + kb) * NHEAD + n];
            bf.y = Wd[(k0 + kb + 1) * NHEAD + n];
            c = __builtin_amdgcn_wmma_f32_16x16x4_f32(
                    false, a, false, bf, (short)0, c, false, false);
        }
    } else {
        const bool mok = (m < NVOC);          // only v==1536 row is real
#pragma unroll
        for (int k0 = 0; k0 < NHEAD; k0 += 4) {
            v2f a, bf;
            float a0 = A[k0 + kb];
            float a1 = A[k0 + kb + 1];
            a.x  = mok ? a0 : 0.f;
            a.y  = mok ? a1 : 0.f;
            bf.x = Wd[(k0 + kb) * NHEAD + n];
            bf.y = Wd[(k0 + kb + 1) * NHEAD + n];
            c = __builtin_amdgcn_wmma_f32_16x16x4_f32(
                    false, a, false, bf, (short)0, c, false, false);
        }
    }

    // C/D layout: VGPR r -> M = 8*(lane>>4) + r ; N = lane & 15.
    const int rowhalf = (lane >> 4) * 8;
    float* dst = table + ((size_t)d * NVOC + (size_t)(vt * 16 + rowhalf)) * NHEAD + n;

    if (full_tile) {
        // 8 stores at immediate offsets from one base -> clauseable b32 stores,
        // each covering two contiguous 64B half-wave segments.
#pragma unroll
        for (int r = 0; r < 8; ++r)
            dst[r * NHEAD] = c[r];
    } else {
        // vt == 96: only v = 1536 (rowhalf==0, r==0) exists.
        if (rowhalf == 0)
            dst[0] = c[0];
    }
}

// -----------------------------------------------------------------------------
// Kernel 2: fused gather-sum + spatial bias + output assembly (interior 64x64).
// Block = 1024 threads = 32 waves; wave w handles cell (b,i,j0+w), lanes = heads
// (coalesced 128B table-row gathers; indices come off a wave-uniform base so
// they scalarize). LDS 32x32 transpose (+pad) gives j-contiguous final stores.
// -----------------------------------------------------------------------------
__global__ __launch_bounds__(1024) void fused_main(
    const float* __restrict__ attn_bias,      // [16][65][65]
    const int*   __restrict__ spatial_pos,    // [16][64][64]
    const int*   __restrict__ edge_input,     // [16][64][64][20][3]
    const float* __restrict__ mask_2d,        // [16]
    const float* __restrict__ spatial_enc_w,  // [512][32]
    const float* __restrict__ table,          // [20][1537][32]
    float* __restrict__ out)                  // [16][32][65][65]
{
    __shared__ float lds[32 * 33];

    const int blk = blockIdx.x;
    const int b   = blk >> 7;                 // / 128
    const int rem = blk & 127;
    const int i   = rem >> 1;
    const int j0  = (rem & 1) << 5;           // 0 or 32

    const int w = __builtin_amdgcn_readfirstlane(threadIdx.x >> 5); // wave id
    const int h = threadIdx.x & 31;           // head = lane
    const int j = j0 + w;

    const float mask = mask_2d[b];

    const int cell  = ((b * NN) + i) * NN + j;
    const int ebase = cell * (NHOP * 3);      // 60 contiguous ints, uniform addr

    float acc = 0.f;
#pragma unroll 5
    for (int d = 0; d < NHOP; ++d) {
        const float* tbl = table + (size_t)d * NVOC * NHEAD + h;
        int i0 = edge_input[ebase + d * 3 + 0];
        int i1 = edge_input[ebase + d * 3 + 1];
        int i2 = edge_input[ebase + d * 3 + 2];
        acc += tbl[i0 * NHEAD] + tbl[i1 * NHEAD] + tbl[i2 * NHEAD];
    }

    const int sp  = spatial_pos[cell];
    int sp_ = (sp == 0) ? 1 : sp;
    sp_ = (sp_ > 1) ? (sp_ - 1) : sp_;
    sp_ = (sp_ < NHOP) ? sp_ : NHOP;

    const float eb  = acc * (1.f / 3.f) / (float)sp_ * mask;
    const float spb = spatial_enc_w[sp * NHEAD + h] * mask;

    lds[w * 33 + h] = spb + eb;               // [j-local][head]; 33-pad -> no conflicts
    __syncthreads();

    // Phase 2: head = w (wave-uniform), jj = lane -> 128B contiguous stores/head.
    const int jj  = threadIdx.x & 31;
    const float v = lds[jj * 33 + w];
    const int col = 1 + j0 + jj;
    const float ab = attn_bias[((b * 65) + (i + 1)) * 65 + col];
    out[(((size_t)b * NHEAD + w) * 65 + (i + 1)) * 65 + col] = 2.f * ab + v;
}

// -----------------------------------------------------------------------------
// Kernel 3: borders. gab[:,:,0,:] += tok ; gab[:,:,1:,0] += tok ; plus the
// doubled attn_bias everywhere. One block per (b,h) plane.
// -----------------------------------------------------------------------------
__global__ __launch_bounds__(256) void borders(
    const float* __restrict__ attn_bias,      // [16][65][65]
    const float* __restrict__ token_w,        // [1][32]
    float* __restrict__ out)                  // [16][32][65][65]
{
    const int b = blockIdx.x >> 5;
    const int h = blockIdx.x & 31;
    const float tok = token_w[h];
    const int t = threadIdx.x;

    float*       plane = out + ((size_t)b * NHEAD + h) * 65 * 65;
    const float* ab    = attn_bias + (size_t)b * 65 * 65;

    if (t < 65) {                             // row 0, all columns
        plane[t] = 2.f * ab[t] + tok;
    } else if (t < 129) {                     // column 0, rows 1..64
        int r = t - 64;
        plane[r * 65] = 2.f * ab[r * 65] + tok;
    }
}

// -----------------------------------------------------------------------------
extern "C" void kernel_launch(void* const* d_in, const int* in_sizes, int n_in,
                              void* d_out, int out_size, void* d_ws, size_t ws_size,
                              hipStream_t stream) {
    const float* attn_bias      = (const float*)d_in[0];
    const int*   spatial_pos    = (const int*)  d_in[1];
    // d_in[2] = x              (unused by reference math)
    const int*   edge_input     = (const int*)  d_in[3];
    // d_in[4] = attn_edge_type (unused by reference math)
    const float* mask_2d        = (const float*)d_in[5];
    const float* edge_enc_w     = (const float*)d_in[6];
    const float* edge_dis_enc_w = (const float*)d_in[7];
    const float* spatial_enc_w  = (const float*)d_in[8];
    const float* token_w        = (const float*)d_in[9];
    float*       out            = (float*)d_out;

    float* table = (float*)d_ws;  // 20*1537*32 floats = 3.94 MB scratch

    // 1) WMMA precompute: table[d] = edge_enc_w @ Wd[d]   (63 MFLOP total)
    build_table_wmma<<<485, 256, 0, stream>>>(edge_enc_w, edge_dis_enc_w, table);

    // 2) fused gather + bias for interior [1:,1:]  (memory-bound, ~25MB HBM)
    fused_main<<<NB * NN * 2, 1024, 0, stream>>>(
        attn_bias, spatial_pos, edge_input, mask_2d, spatial_enc_w, table, out);

    // 3) borders (row 0 / col 0 token bias)
    borders<<<NB * NHEAD, 256, 0, stream>>>(attn_bias, token_w, out);
}